// LongConvBlock_58222576665045
// MI455X (gfx1250) — compile-verified
//
#include <hip/hip_runtime.h>

typedef __attribute__((ext_vector_type(2))) float v2f;
typedef __attribute__((ext_vector_type(4))) float v4f;
typedef __attribute__((ext_vector_type(8))) float v8f;

constexpr int SEQ  = 4096;   // L
constexpr int DIM  = 1024;   // H (channels)
constexpr int NB   = 4;      // batch
constexpr int KPAD = 16;     // causal zero pad (t >= -15)
constexpr int BPAD = 15;     // leading zero x-blocks (block index >= -15)
constexpr int XSTR = 20;     // x block stride in floats: 20 banks -> conflict-free
// kq[4u+0..3] = (kth[u-KPAD], kth[u-KPAD-1], kth[u-KPAD-4], kth[u-KPAD-5])
constexpr int KQ_SIZE = 4 * (KPAD + SEQ + 5) + 3;            // max write 16467
constexpr int XP_SIZE = XSTR * (BPAD + SEQ / 16);            // 271 blocks

__device__ __forceinline__ int pperm(int c) {   // {0,1,4,5,8,9,12,13|2,3,...}
  return ((c >> 1) & 1) * 8 + ((c >> 2) << 1) + (c & 1);
}

// One workgroup = one (b,h) sequence. 8 waves; wave w computes output
// column-tiles {w, 15-w} so every wave runs exactly 272 d-iterations.
__global__ __launch_bounds__(256)
void longconv_wmma_f32(const float* __restrict__ x,
                       const float* __restrict__ kern,
                       const float* __restrict__ dskip,
                       float* __restrict__ out) {
  __shared__ alignas(16) float kq[KQ_SIZE];  // quad-interleaved kernel
  __shared__ alignas(16) float xp[XP_SIZE];  // permuted x blocks, stride 20

  const int h   = blockIdx.x;
  const int b   = blockIdx.y;
  const int tid = threadIdx.x;

  // Zero pad x blocks 0..14 (only the 16 data slots per block are ever read).
  if (tid < 240) xp[XSTR * (tid >> 4) + (tid & 15)] = 0.0f;
  // Zero exactly the kq slots whose source kth index is negative (race-free:
  // these slots are never written by the staging loop below).
  if (tid < 88) {
    const int u = tid >> 2, r = tid & 3;
    const int off = (r == 0) ? 0 : (r == 1) ? 1 : (r == 2) ? 4 : 5;
    if (u - KPAD - off < 0) kq[tid] = 0.0f;
  }

  // Kernel loads fully coalesced; x loads strided by DIM but the working set
  // (144 MB) is resident in MI455X's 192 MB L2.
  const float* kh_ptr = kern + (size_t)h * SEQ;
  const float* xh_ptr = x + (size_t)b * SEQ * DIM + h;
  #pragma unroll
  for (int j = 0; j < SEQ / 256; ++j) {
    const int i = tid + 256 * j;
    const float kv  = kh_ptr[i];
    const float mag = fabsf(kv) - 0.1f;               // relu(|k|-0.1)*sign(k)
    const float kt  = mag > 0.0f ? copysignf(mag, kv) : 0.0f;
    const int kb = 4 * (KPAD + i);
    kq[kb]      = kt;   // slot (u=K+i,   r=0)
    kq[kb + 5]  = kt;   // slot (u=K+i+1, r=1)
    kq[kb + 18] = kt;   // slot (u=K+i+4, r=2)
    kq[kb + 23] = kt;   // slot (u=K+i+5, r=3)
    xp[XSTR * (BPAD + (i >> 4)) + pperm(i & 15)] = xh_ptr[(size_t)i * DIM];
  }
  __syncthreads();

  const int lane = tid & 31;
  const int wave = tid >> 5;
  const int n    = lane & 15;   // A row (M) and B column (N) for this lane
  const int half = lane >> 4;   // 0: K slots {0,1}, 1: K slots {2,3}
  const int kh2  = half * 2;

  const float dsk = dskip[h];

  for (int rep = 0; rep < 2; ++rep) {
    const int ci = (rep == 0) ? wave : 15 - wave;   // output column-tile index
    v8f acc = {};                                   // 16x16 f32 accumulator

    // A: b128 at 4*(KPAD + 16d + n - kh2) -> parts p=0,1; at -32 -> p=2,3.
    // B: b128 at XSTR*(BPAD + 16ci + n - d) + 8*half (+4) -> p=0,1 (p=2,3).
    const int a_base = 4 * (KPAD + n - kh2);                    // + 64*d
    const int b_base = XSTR * (BPAD + 16 * ci + n) + 8 * half;  // - XSTR*d

    const int dmax = 16 * ci + 16;      // causality: only d <= output block idx
    for (int d = 0; d < dmax; ++d) {
      const int sa = a_base + 64 * d;
      const int sb = b_base - XSTR * d;
      const v4f aq0 = *(const v4f*)(kq + sa);
      const v4f aq1 = *(const v4f*)(kq + sa - 32);
      const v4f bq0 = *(const v4f*)(xp + sb);
      const v4f bq1 = *(const v4f*)(xp + sb + 4);
      acc = __builtin_amdgcn_wmma_f32_16x16x4_f32(
          false, __builtin_shufflevector(aq0, aq0, 0, 1),
          false, __builtin_shufflevector(bq0, bq0, 0, 1), (short)0, acc, false, false);
      acc = __builtin_amdgcn_wmma_f32_16x16x4_f32(
          false, __builtin_shufflevector(aq0, aq0, 2, 3),
          false, __builtin_shufflevector(bq0, bq0, 2, 3), (short)0, acc, false, false);
      acc = __builtin_amdgcn_wmma_f32_16x16x4_f32(
          false, __builtin_shufflevector(aq1, aq1, 0, 1),
          false, __builtin_shufflevector(bq1, bq1, 0, 1), (short)0, acc, false, false);
      acc = __builtin_amdgcn_wmma_f32_16x16x4_f32(
          false, __builtin_shufflevector(aq1, aq1, 2, 3),
          false, __builtin_shufflevector(bq1, bq1, 2, 3), (short)0, acc, false, false);
    }

    // C/D layout: vgpr v -> row m = v + 8*half, lane n -> block column 16ci+n.
    #pragma unroll
    for (int v = 0; v < 8; ++v) {
      const int c  = v + 8 * half;                       // position within block
      const int jb = 16 * ci + n;                        // block index
      const int l  = jb * 16 + c;                        // sequence position
      const float xv = xp[XSTR * (BPAD + jb) + pperm(c)];
      out[((size_t)b * SEQ + l) * DIM + h] = acc[v] + dsk * xv;  // skip term
    }
  }
}

extern "C" void kernel_launch(void* const* d_in, const int* in_sizes, int n_in,
                              void* d_out, int out_size, void* d_ws, size_t ws_size,
                              hipStream_t stream) {
  const float* x    = (const float*)d_in[0];  // [4, 4096, 1024]
  const float* kern = (const float*)d_in[1];  // [1, 1024, 4096]
  const float* dsk  = (const float*)d_in[2];  // [1, 1024]
  float* out        = (float*)d_out;          // [4, 4096, 1024]

  dim3 grid(DIM, NB);   // 4096 workgroups: one per (b,h) sequence
  longconv_wmma_f32<<<grid, 256, 0, stream>>>(x, kern, dsk, out);
}